// EPAll2AllLayer_15496242004360
// MI455X (gfx1250) — compile-verified
//
#include <hip/hip_runtime.h>
#include <hip/hip_bf16.h>

// EP all-to-all dispatch+combine, single process.
// Key identity: combine applies the inverse of the dispatch permutation, so
//   gathered[t,k,:] == input[t,:]  for every k, and
//   out[t,h] = (sum_k weights[t,k]) * input[t,h].
// exp_indices cancels out entirely. The kernel is pure streaming:
// ~128 MiB moved, ~0.125 flop/byte -> HBM-bound (floor ~5.6 us @ 23.3 TB/s).
//
// Data path: gfx1250 async global->LDS pipeline. Each thread owns private
// 32B slots in a 4-deep LDS ring; GLOBAL_LOAD_ASYNC_TO_LDS_B128 prefetches
// 3 rows ahead (ASYNCcnt-throttled), DS_LOAD_B128 pulls into VGPRs, one
// v_mul per element, GLOBAL_STORE_B128 out. No barriers needed: every lane
// reads back exactly the LDS bytes it fetched itself.

#define T_TOK 8192
#define H_DIM 2048
#define K_EXP 8
#define TPB   256   // threads per block: 256 * 32B = 8 KiB = one row
#define RPB   8     // rows per block
#define NBUF  4     // LDS ring depth (32 KiB total)

__global__ __launch_bounds__(TPB) void ep_combine_scale_kernel(
    const float* __restrict__ x,     // [T, H]
    const float* __restrict__ w,     // [T, K]
    float* __restrict__ out)         // [T, H]
{
    __shared__ float4 sbuf[NBUF][TPB * 2];   // 4 x 8 KiB ring

    const int tid  = threadIdx.x;
    const int base = blockIdx.x * RPB;

    // LDS byte offsets of this thread's two 16B slots in each ring buffer.
    // (flat shared address low 32 bits == workgroup-relative LDS offset)
    unsigned lds_off[NBUF];
#pragma unroll
    for (int b = 0; b < NBUF; ++b)
        lds_off[b] = (unsigned)(uintptr_t)(&sbuf[b][tid * 2]);

    // Issue the async fetch of row r into ring slot `lo`.
    // s_wait_dscnt 0 first: guarantees any previous ds_load of this buffer
    // has drained before the async engine overwrites it (WAR on reuse).
    auto issue = [&](int r, unsigned lo) {
        const float* gp = x + (size_t)(base + r) * H_DIM + tid * 8; // 8 floats = 32B
        unsigned long long ga = (unsigned long long)(uintptr_t)gp;
        asm volatile(
            "s_wait_dscnt 0\n\t"
            "global_load_async_to_lds_b128 %0, %1, off\n\t"
            "global_load_async_to_lds_b128 %0, %1, off offset:16"
            :
            : "v"(lo), "v"(ga)
            : "memory");
    };

    // Prime the pipeline: rows 0..2 in flight (2 async ops per row).
    issue(0, lds_off[0]);
    issue(1, lds_off[1]);
    issue(2, lds_off[2]);

#pragma unroll
    for (int r = 0; r < RPB; ++r) {
        // Keep 3 rows of prefetch in flight.
        if (r + 3 < RPB) issue(r + 3, lds_off[(r + 3) & (NBUF - 1)]);

        // Row weight sum (wave-uniform address -> scalar loads, L2-resident).
        const float* wr = w + (size_t)(base + r) * K_EXP;
        float ws = 0.0f;
#pragma unroll
        for (int k = 0; k < K_EXP; ++k) ws += wr[k];

        // Async loads complete in order: allow only rows > r to be pending.
        // rows still in flight after wait = min(r+3, RPB-1) - r, x2 ops each.
        if      (r + 3 < RPB) asm volatile("s_wait_asynccnt 6" ::: "memory");
        else if (r + 2 < RPB) asm volatile("s_wait_asynccnt 4" ::: "memory");
        else if (r + 1 < RPB) asm volatile("s_wait_asynccnt 2" ::: "memory");
        else                  asm volatile("s_wait_asynccnt 0" ::: "memory");

        // Pull this thread's 32B back from LDS, scale, store 2 x b128.
        const float4* sp = &sbuf[r & (NBUF - 1)][tid * 2];
        float4 a = sp[0];
        float4 b = sp[1];
        float4 ra = make_float4(a.x * ws, a.y * ws, a.z * ws, a.w * ws);
        float4 rb = make_float4(b.x * ws, b.y * ws, b.z * ws, b.w * ws);

        float4* op = (float4*)out + (size_t)(base + r) * (H_DIM / 4) + tid * 2;
        op[0] = ra;
        op[1] = rb;
    }
}

extern "C" void kernel_launch(void* const* d_in, const int* in_sizes, int n_in,
                              void* d_out, int out_size, void* d_ws, size_t ws_size,
                              hipStream_t stream) {
    const float* x = (const float*)d_in[0];   // input   [T, H] f32
    const float* w = (const float*)d_in[1];   // weights [T, K] f32
    // d_in[2] (exp_indices) provably does not affect the output: the combine
    // applies the exact inverse of the dispatch permutation.
    float* out = (float*)d_out;               // [T, H] f32

    dim3 grid(T_TOK / RPB);   // 1024 blocks
    dim3 block(TPB);          // 256 threads = 8 wave32
    ep_combine_scale_kernel<<<grid, block, 0, stream>>>(x, w, out);
}